// SLMGAE_79405355369094
// MI455X (gfx1250) — compile-verified
//
#include <hip/hip_runtime.h>

typedef __attribute__((ext_vector_type(2))) float v2f;
typedef __attribute__((ext_vector_type(8))) float v8f;

#define N_NODES 20000
#define N_EDGES 500000

#define H1S 404  // h1 row stride (pad 400 -> 404: conflict-free ds_load_b64)
#define H2S 212  // h2 row stride (pad 208 -> 212: conflict-free ds_load_b64)

// ---------------------------------------------------------------------------
// Kernel 0: per-node MLP 4 -> 64 -> 32 -> 16, writes m into zc[:, 64:80]
// ---------------------------------------------------------------------------
__global__ void k_node_mlp(const float* __restrict__ cell,
                           const float* __restrict__ w1, const float* __restrict__ b1,
                           const float* __restrict__ w2, const float* __restrict__ b2,
                           const float* __restrict__ w3, const float* __restrict__ b3,
                           float* __restrict__ zc) {
  int node = blockIdx.x * blockDim.x + threadIdx.x;
  if (node >= N_NODES) return;
  float c0 = cell[node * 4 + 0], c1 = cell[node * 4 + 1];
  float c2 = cell[node * 4 + 2], c3 = cell[node * 4 + 3];
  float t1[64];
#pragma unroll 8
  for (int j = 0; j < 64; ++j) {
    float s = b1[j] + c0 * w1[0 * 64 + j] + c1 * w1[1 * 64 + j] +
              c2 * w1[2 * 64 + j] + c3 * w1[3 * 64 + j];
    t1[j] = s > 0.f ? s : 0.f;
  }
  float t2[32];
  for (int j = 0; j < 32; ++j) {
    float s = b2[j];
#pragma unroll 8
    for (int i = 0; i < 64; ++i) s += t1[i] * w2[i * 32 + j];
    t2[j] = s > 0.f ? s : 0.f;
  }
  for (int j = 0; j < 16; ++j) {
    float s = b3[j];
#pragma unroll 8
    for (int i = 0; i < 32; ++i) s += t2[i] * w3[i * 16 + j];
    zc[(long long)node * 400 + 64 + j] = s;
  }
}

// ---------------------------------------------------------------------------
// Kernel 1: fill zc[:,0:64] = z, zc[:,80:400] = esm  (float4 vectorized)
// ---------------------------------------------------------------------------
__global__ void k_zc_fill(const float4* __restrict__ z4,
                          const float4* __restrict__ esm4,
                          float* __restrict__ zc) {
  int id = blockIdx.x * blockDim.x + threadIdx.x;
  if (id >= N_NODES * 96) return;  // 96 float4 per node (16 z + 80 esm)
  int node = id / 96;
  int c4 = id - node * 96;
  float4 v;
  long long dst;
  if (c4 < 16) {
    v = z4[(long long)node * 16 + c4];
    dst = (long long)node * 400 + c4 * 4;
  } else {
    v = esm4[(long long)node * 80 + (c4 - 16)];
    dst = (long long)node * 400 + 80 + (c4 - 16) * 4;
  }
  *(float4*)(zc + dst) = v;
}

// ---------------------------------------------------------------------------
// Weight packer: W[K][N] (row-major) -> per-lane WMMA f32 16x16x4 B fragments.
// packed[((ct*(Kp/4) + s)*32 + lane)*2 + j] = W[4s + 2*(lane>>4) + j][ct*16 + (lane&15)]
// (zero-filled outside [K,N]).  One thread per packed float.
// ---------------------------------------------------------------------------
__global__ void k_pack_b(const float* __restrict__ W, float* __restrict__ packed,
                         int K, int N, int Kp4, int total) {
  int id = blockIdx.x * blockDim.x + threadIdx.x;
  if (id >= total) return;
  int j = id & 1;
  int lane = (id >> 1) & 31;
  int rem = id >> 6;  // ct*Kp4 + s
  int s = rem % Kp4;
  int ct = rem / Kp4;
  int k = 4 * s + 2 * (lane >> 4) + j;
  int n = ct * 16 + (lane & 15);
  packed[id] = (k < K && n < N) ? W[(long long)k * N + n] : 0.f;
}

// ---------------------------------------------------------------------------
// Kernel 2: P[N,800]: P[:,0:400] = zc @ fc1_w[0:400,:], P[:,400:800] = zc @ fc1_w[400:800,:]
// One wave per 16x16 output tile, K=400 via V_WMMA_F32_16X16X4_F32, packed B.
// ---------------------------------------------------------------------------
__global__ __launch_bounds__(128) void k_proj_gemm(const float* __restrict__ zc,
                                                   const float* __restrict__ pw1,
                                                   float* __restrict__ P) {
  int wid = blockIdx.x * 4 + (threadIdx.x >> 5);  // 0 .. 62499
  int lane = threadIdx.x & 31;
  int r = wid / 50;      // node tile 0..1249
  int c = wid % 50;      // output col tile 0..49
  int kslab = c / 25;    // 0: fc1_w rows 0..399, 1: rows 400..799
  int cc = c % 25;       // 16-col tile within slab
  int n = lane & 15;     // A: M index, B: N index
  int half = lane >> 4;  // K half

  const float* arow = zc + (long long)(r * 16 + n) * 400 + 2 * half;
  // packed fc1: Ntiles=25, Kp4=200; fragment (ct=cc, sGlobal=kslab*100+s)
  const float* bfrag = pw1 + ((long long)cc * 200 + kslab * 100) * 64 + lane * 2;

  v8f acc = {};
  for (int s = 0; s < 100; ++s) {
    v2f a = *(const v2f*)(arow + 4 * s);
    v2f b = *(const v2f*)(bfrag + s * 64);
    acc = __builtin_amdgcn_wmma_f32_16x16x4_f32(false, a, false, b, (short)0, acc,
                                                false, false);
  }
  float* prow = P + (long long)(r * 16) * 800 + kslab * 400 + cc * 16 + n;
#pragma unroll
  for (int v = 0; v < 8; ++v) {
    int orow = v + 8 * half;
    prow[(long long)orow * 800] = acc[v];
  }
}

// ---------------------------------------------------------------------------
// Kernel 3: per 16-edge tile: h1 = relu(P[src,0:400] + P[dst,400:800] + b1),
// then fc2 (400->200) and fc3 (200->100) via WMMA f32 with packed B,
// fc4 (100->1) split over 32 lanes.
// ---------------------------------------------------------------------------
__global__ __launch_bounds__(128) void k_edge_mlp(
    const long long* __restrict__ ei, const float* __restrict__ P,
    const float* __restrict__ b1, const float* __restrict__ pw2,
    const float* __restrict__ b2, const float* __restrict__ pw3,
    const float* __restrict__ b3, const float* __restrict__ w4,
    const float* __restrict__ b4, float* __restrict__ out) {
  __shared__ float h1[16 * H1S];
  __shared__ float h2[16 * H2S];  // 200 cols padded to 208 (13 WMMA col tiles)
  __shared__ float h3[16 * 112];  // 100 cols padded to 112 (7 WMMA col tiles)
  __shared__ int s_src[16];
  __shared__ int s_dst[16];
  __shared__ float psum[32];

  int tid = threadIdx.x;
  int ebase = blockIdx.x * 16;

  if (tid < 16) {
    s_src[tid] = (int)ei[ebase + tid];
    s_dst[tid] = (int)ei[N_EDGES + ebase + tid];
  }
  __syncthreads();

  // h1 in LDS: float4-vectorized gather of two 400-float rows of P per edge.
  // 16 edges * 100 float4 = 1600 float4 elements over 128 threads.
  for (int id = tid; id < 1600; id += 128) {
    int m = id / 100;
    int j4 = id - m * 100;
    float4 a = ((const float4*)(P + (long long)s_src[m] * 800))[j4];
    float4 b = ((const float4*)(P + (long long)s_dst[m] * 800 + 400))[j4];
    float4 c = ((const float4*)b1)[j4];
    float4 r;
    r.x = fmaxf(a.x + b.x + c.x, 0.f);
    r.y = fmaxf(a.y + b.y + c.y, 0.f);
    r.z = fmaxf(a.z + b.z + c.z, 0.f);
    r.w = fmaxf(a.w + b.w + c.w, 0.f);
    ((float4*)(h1 + m * H1S))[j4] = r;  // H1S*4 = 1616 bytes, 16B aligned
  }
  __syncthreads();

  int wave = tid >> 5;
  int lane = tid & 31;
  int n = lane & 15;
  int half = lane >> 4;

  // fc2: [16,400] @ [400,200] -> h2, 13 col tiles over 4 waves (Kp4 = 100)
  for (int ct = wave; ct < 13; ct += 4) {
    const float* bfrag = pw2 + (long long)ct * 100 * 64 + lane * 2;
    v8f acc = {};
    for (int s = 0; s < 100; ++s) {
      v2f a = *(const v2f*)&h1[n * H1S + 4 * s];
      v2f b = *(const v2f*)(bfrag + s * 64);
      acc = __builtin_amdgcn_wmma_f32_16x16x4_f32(false, a, false, b, (short)0, acc,
                                                  false, false);
    }
    int col = ct * 16 + n;
#pragma unroll
    for (int v = 0; v < 8; ++v) {
      int row = v + 8 * half;
      float val = 0.f;
      if (col < 200) {
        float t = acc[v] + b2[col];
        val = t > 0.f ? t : 0.f;
      }
      h2[row * H2S + col] = val;  // pad cols 200..207 set to 0 (never read as K)
    }
  }
  __syncthreads();

  // fc3: [16,200] @ [200,100] -> h3, 7 col tiles over 4 waves (Kp4 = 50)
  for (int ct = wave; ct < 7; ct += 4) {
    const float* bfrag = pw3 + (long long)ct * 50 * 64 + lane * 2;
    v8f acc = {};
    for (int s = 0; s < 50; ++s) {
      v2f a = *(const v2f*)&h2[n * H2S + 4 * s];
      v2f b = *(const v2f*)(bfrag + s * 64);
      acc = __builtin_amdgcn_wmma_f32_16x16x4_f32(false, a, false, b, (short)0, acc,
                                                  false, false);
    }
    int col = ct * 16 + n;
#pragma unroll
    for (int v = 0; v < 8; ++v) {
      int row = v + 8 * half;
      float val = 0.f;
      if (col < 100) {
        float t = acc[v] + b3[col];
        val = t > 0.f ? t : 0.f;
      }
      h3[row * 112 + col] = val;
    }
  }
  __syncthreads();

  // fc4: 100 -> 1, split each edge's dot over 2 lanes (50 MACs each)
  if (tid < 32) {
    int m = tid & 15;
    int h = tid >> 4;
    float s = 0.f;
    int k0 = h * 50;
#pragma unroll 5
    for (int k = k0; k < k0 + 50; ++k) s += h3[m * 112 + k] * w4[k];
    psum[tid] = s;
  }
  __syncthreads();
  if (tid < 16) out[ebase + tid] = psum[tid] + psum[tid + 16] + b4[0];
}

// ---------------------------------------------------------------------------
extern "C" void kernel_launch(void* const* d_in, const int* in_sizes, int n_in,
                              void* d_out, int out_size, void* d_ws, size_t ws_size,
                              hipStream_t stream) {
  const float* z = (const float*)d_in[0];
  const long long* ei = (const long long*)d_in[1];  // int64 [2, E]
  const float* cell = (const float*)d_in[2];
  const float* esm = (const float*)d_in[3];
  const float* mw1 = (const float*)d_in[4];
  const float* mb1 = (const float*)d_in[5];
  const float* mw2 = (const float*)d_in[6];
  const float* mb2 = (const float*)d_in[7];
  const float* mw3 = (const float*)d_in[8];
  const float* mb3 = (const float*)d_in[9];
  const float* fc1w = (const float*)d_in[10];
  const float* fc1b = (const float*)d_in[11];
  const float* fc2w = (const float*)d_in[12];
  const float* fc2b = (const float*)d_in[13];
  const float* fc3w = (const float*)d_in[14];
  const float* fc3b = (const float*)d_in[15];
  const float* fc4w = (const float*)d_in[16];
  const float* fc4b = (const float*)d_in[17];
  float* out = (float*)d_out;

  // Workspace layout (floats):
  //   zc  [N,400]                 : 8,000,000
  //   P   [N,800]                 : 16,000,000
  //   pw1 (fc1 packed, 25ct*200s) : 320,000
  //   pw2 (fc2 packed, 13ct*100s) : 83,200
  //   pw3 (fc3 packed,  7ct* 50s) : 22,400
  float* zc = (float*)d_ws;
  float* P = zc + (long long)N_NODES * 400;
  float* pw1 = P + (long long)N_NODES * 800;
  float* pw2 = pw1 + 320000;
  float* pw3 = pw2 + 83200;

  // Pack weights into WMMA B-fragment layout (tiny, L2-resident).
  k_pack_b<<<(320000 + 255) / 256, 256, 0, stream>>>(fc1w, pw1, 800, 400, 200, 320000);
  k_pack_b<<<(83200 + 255) / 256, 256, 0, stream>>>(fc2w, pw2, 400, 200, 100, 83200);
  k_pack_b<<<(22400 + 255) / 256, 256, 0, stream>>>(fc3w, pw3, 200, 100, 50, 22400);

  k_node_mlp<<<(N_NODES + 255) / 256, 256, 0, stream>>>(cell, mw1, mb1, mw2, mb2,
                                                        mw3, mb3, zc);
  k_zc_fill<<<(N_NODES * 96 + 255) / 256, 256, 0, stream>>>(
      (const float4*)z, (const float4*)esm, zc);

  // 1250 row tiles * 50 col tiles = 62500 waves, 4 waves/block
  k_proj_gemm<<<15625, 128, 0, stream>>>(zc, pw1, P);

  // 500000 / 16 = 31250 edge tiles
  k_edge_mlp<<<31250, 128, 0, stream>>>(ei, P, fc1b, pw2, fc2b, pw3, fc3b, fc4w,
                                        fc4b, out);
}